// H2GCN_78176994721833
// MI455X (gfx1250) — compile-verified
//
#include <hip/hip_runtime.h>
#include <hip/hip_bf16.h>

typedef __attribute__((ext_vector_type(2))) float v2f;
typedef __attribute__((ext_vector_type(8))) float v8f;

__device__ __forceinline__ void atomAddG(float* p, float v) {
  __hip_atomic_fetch_add(p, v, __ATOMIC_RELAXED, __HIP_MEMORY_SCOPE_AGENT);
}
__device__ __forceinline__ void atomAddL(float* p, float v) {
  __hip_atomic_fetch_add(p, v, __ATOMIC_RELAXED, __HIP_MEMORY_SCOPE_WORKGROUP);
}

// ---------------- utility: zero a float region (grid-stride) ----------------
__global__ void zero_f32_kernel(float* __restrict__ p, size_t n) {
  size_t i = (size_t)blockIdx.x * blockDim.x + threadIdx.x;
  size_t stride = (size_t)gridDim.x * blockDim.x;
  for (; i < n; i += stride) p[i] = 0.0f;
}

// ---------------- graph preprocessing ----------------
// Each raw edge (u,v) represents both directed edges (u,v) and (v,u).
__global__ void deg_kernel(const int* __restrict__ ei, int E, float* __restrict__ deg) {
  int e = blockIdx.x * blockDim.x + threadIdx.x;
  if (e >= E) return;
  int u = ei[e], v = ei[E + e];
  if (u != v) {
    atomAddG(&deg[u], 1.0f);
    atomAddG(&deg[v], 1.0f);
  }
}

__global__ void dinv_kernel(float* __restrict__ deg, int N) {
  int i = blockIdx.x * blockDim.x + threadIdx.x;
  if (i >= N) return;
  float d = deg[i];
  deg[i] = (d > 0.0f) ? rsqrtf(d) : 0.0f;  // in place: deg -> d^-1/2
}

__global__ void wdiag_kernel(const int* __restrict__ ei, int E,
                             const float* __restrict__ dinv,
                             float* __restrict__ wbuf, float* __restrict__ diag2) {
  int e = blockIdx.x * blockDim.x + threadIdx.x;
  if (e >= E) return;
  int u = ei[e], v = ei[E + e];
  float w = 0.0f;
  if (u != v) w = dinv[u] * dinv[v];
  wbuf[e] = w;
  if (w != 0.0f) {
    float w2 = w * w;           // symmetric: same w^2 lands on row u and row v
    atomAddG(&diag2[u], w2);
    atomAddG(&diag2[v], w2);
  }
}

// ------- fp32 WMMA linear: out = act(A[MxK] @ W[OxK]^T + b), TM x TN tiles/wave -------
// Per wave: TM row-tiles x TN col-tiles (TN covers ALL output columns), so A is
// streamed from HBM exactly once and B fragments live in registers for 16*TM rows.
// Fragment layouts per ISA 7.12.2 (wave32):
//   A 16x4 : lane l holds A[l&15][k + 2*(l>>4)], A[...+1]
//   B 4x16 : lane l holds B[k + 2*(l>>4)][l&15] = W[col][k + 2*(l>>4)], ...
//   C 16x16: vgpr r -> rows r (lanes 0-15) and r+8 (lanes 16-31)
template <int TM, int TN>
__global__ void wmma_lin_kernel(const float* __restrict__ A, int lda,
                                const float* __restrict__ W, int ldw,
                                const float* __restrict__ bias,
                                float* __restrict__ out, int ldo,
                                int nrows, int row_blocks,
                                int K, int do_relu, int ncols) {
  int wid = blockIdx.x * (blockDim.x >> 5) + (threadIdx.x >> 5);
  if (wid >= row_blocks) return;              // wave-uniform: EXEC stays all-ones
  int l = threadIdx.x & 31;
  int ml = l & 15;
  int half = l >> 4;
  int m0 = wid * (TM * 16);

  const float* arow[TM];
#pragma unroll
  for (int i = 0; i < TM; ++i) {
    int r = m0 + i * 16 + ml;
    if (r >= nrows) r = nrows - 1;            // clamp loads; stores guarded below
    arow[i] = A + (size_t)r * lda + 2 * half;
  }
  const float* wrow[TN];
#pragma unroll
  for (int j = 0; j < TN; ++j) {
    int c = j * 16 + ml;
    if (c >= ncols) c = 0;                    // clamp loads for padded cols
    wrow[j] = W + (size_t)c * ldw + 2 * half;
  }

  v8f acc[TM][TN];
#pragma unroll
  for (int i = 0; i < TM; ++i)
#pragma unroll
    for (int j = 0; j < TN; ++j) acc[i][j] = {};

#pragma unroll 2
  for (int k = 0; k < K; k += 4) {
    v2f a[TM], b[TN];
#pragma unroll
    for (int j = 0; j < TN; ++j) { b[j].x = wrow[j][k]; b[j].y = wrow[j][k + 1]; }
#pragma unroll
    for (int i = 0; i < TM; ++i) { a[i].x = arow[i][k]; a[i].y = arow[i][k + 1]; }
#pragma unroll
    for (int i = 0; i < TM; ++i)
#pragma unroll
      for (int j = 0; j < TN; ++j)
        acc[i][j] = __builtin_amdgcn_wmma_f32_16x16x4_f32(
            /*neg_a=*/false, a[i], /*neg_b=*/false, b[j],
            /*c_mod=*/(short)0, acc[i][j], /*reuse_a=*/false, /*reuse_b=*/false);
  }

#pragma unroll
  for (int j = 0; j < TN; ++j) {
    int c = j * 16 + ml;
    float bv = (c < ncols) ? bias[c] : 0.0f;
#pragma unroll
    for (int i = 0; i < TM; ++i) {
#pragma unroll
      for (int r = 0; r < 8; ++r) {
        int m = m0 + i * 16 + r + 8 * half;
        float vv = acc[i][j][r] + bv;
        if (do_relu) vv = fmaxf(vv, 0.0f);
        if (c < ncols && m < nrows) out[(size_t)m * ldo + c] = vv;
      }
    }
  }
}

// ---------------- BatchNorm ----------------
__global__ void bn_stats_kernel(const float* __restrict__ h, int N,
                                float* __restrict__ gsum, float* __restrict__ gsumsq) {
  __shared__ float ls[64];
  __shared__ float ls2[64];
  int t = threadIdx.x;
  int f = t & 63;
  int g = t >> 6;                                   // 4 row-groups per block
  if (t < 64) { ls[t] = 0.0f; ls2[t] = 0.0f; }
  __syncthreads();
  float s = 0.0f, s2 = 0.0f;
  for (int r = blockIdx.x * 4 + g; r < N; r += gridDim.x * 4) {
    float v = h[(size_t)r * 64 + f];
    s += v; s2 += v * v;
  }
  atomAddL(&ls[f], s);
  atomAddL(&ls2[f], s2);
  __syncthreads();
  if (t < 64) {
    atomAddG(&gsum[t], ls[t]);
    atomAddG(&gsumsq[t], ls2[t]);
  }
}

__global__ void bn_final_kernel(const float* __restrict__ gsum, const float* __restrict__ gsumsq,
                                const float* __restrict__ gamma, const float* __restrict__ beta,
                                int N, float* __restrict__ scsh) {
  int f = threadIdx.x;
  if (f >= 64) return;
  float inv_n = 1.0f / (float)N;
  float mu = gsum[f] * inv_n;
  float var = gsumsq[f] * inv_n - mu * mu;          // biased variance
  float sc = gamma[f] * rsqrtf(var + 1e-5f);
  scsh[f] = sc;
  scsh[64 + f] = beta[f] - mu * sc;
}

// writes normalized h into z[:,0:64] and into cur
__global__ void bn_apply_kernel(const float* __restrict__ h, const float* __restrict__ scsh,
                                float* __restrict__ z, float* __restrict__ cur, int N) {
  size_t i = (size_t)blockIdx.x * blockDim.x + threadIdx.x;
  if (i >= (size_t)N * 64) return;
  int f = (int)(i & 63);
  size_t n = i >> 6;
  float v = h[i] * scsh[f] + scsh[64 + f];
  z[n * 320 + f] = v;
  cur[i] = v;
}

// ---------------- SpMM: y += D^-1/2 A D^-1/2 @ x (scatter both directions) --------
// one wave per raw edge; lane handles feature pair (2*lane, 2*lane+1)
__global__ void spmm_kernel(const int* __restrict__ ei, const float* __restrict__ wbuf, int E,
                            const float* __restrict__ xin, int ldi,
                            float* __restrict__ yout, int ldo) {
  int e = blockIdx.x * (blockDim.x >> 5) + (threadIdx.x >> 5);
  if (e >= E) return;
  float w = wbuf[e];
  if (w == 0.0f) return;                       // self loop (wave-uniform)
  int lane = threadIdx.x & 31;
  int u = ei[e], v = ei[E + e];
  int f = lane * 2;
  const float* xu = xin + (size_t)u * ldi + f;
  const float* xv = xin + (size_t)v * ldi + f;
  float a0 = xu[0], a1 = xu[1];
  float b0 = xv[0], b1 = xv[1];
  float* yu = yout + (size_t)u * ldo + f;
  float* yv = yout + (size_t)v * ldo + f;
  atomAddG(&yu[0], w * b0);
  atomAddG(&yu[1], w * b1);
  atomAddG(&yv[0], w * a0);
  atomAddG(&yv[1], w * a1);
}

// h2 = (A @ h1)_acc - diag2 * cur ; cur <- h1 + h2    (slots inside z[N,320])
__global__ void h2fix_kernel(float* __restrict__ z, float* __restrict__ cur,
                             const float* __restrict__ diag2,
                             int off1, int off2, int N) {
  size_t i = (size_t)blockIdx.x * blockDim.x + threadIdx.x;
  if (i >= (size_t)N * 64) return;
  int f = (int)(i & 63);
  size_t n = i >> 6;
  float h1 = z[n * 320 + off1 + f];
  float h2 = z[n * 320 + off2 + f] - diag2[n] * cur[i];
  z[n * 320 + off2 + f] = h2;
  cur[i] = h1 + h2;
}

// ---------------- host orchestration ----------------
static inline int cdiv(long a, long b) { return (int)((a + b - 1) / b); }

extern "C" void kernel_launch(void* const* d_in, const int* in_sizes, int n_in,
                              void* d_out, int out_size, void* d_ws, size_t ws_size,
                              hipStream_t stream) {
  const float* x     = (const float*)d_in[0];
  const int*   ei    = (const int*)  d_in[1];
  const float* Wemb  = (const float*)d_in[2];
  const float* bemb  = (const float*)d_in[3];
  const float* gamma = (const float*)d_in[4];
  const float* beta  = (const float*)d_in[5];
  const float* W1    = (const float*)d_in[6];
  const float* b1    = (const float*)d_in[7];
  const float* W2    = (const float*)d_in[8];
  const float* b2    = (const float*)d_in[9];

  const int IN = 512, H = 64;
  const int N = in_sizes[0] / IN;       // 100000
  const int E = in_sizes[1] / 2;        // 800000
  const int C = out_size / N;           // 40

  // workspace layout (floats); accumulation regions first so one zero pass covers them
  float* wsf   = (float*)d_ws;
  size_t o = 0;
  float* deg   = wsf + o; o += (size_t)N;        // then holds d^-1/2
  float* diag2 = wsf + o; o += (size_t)N;
  float* sums  = wsf + o; o += 64;
  float* sumsq = wsf + o; o += 64;
  size_t zero_head = o;                          // deg,diag2,sums,sumsq
  float* scsh  = wsf + o; o += 128;
  float* wbuf  = wsf + o; o += (size_t)E;
  float* hbuf  = wsf + o; o += (size_t)N * H;    // pre-BN embed
  float* cur   = wsf + o; o += (size_t)N * H;    // cur; reused as g after hops
  float* z     = wsf + o; o += (size_t)N * 5 * H; // concat [N,320]

  const int T = 256;

  // zero accumulators
  zero_f32_kernel<<<1024, T, 0, stream>>>(wsf, zero_head);
  zero_f32_kernel<<<4096, T, 0, stream>>>(z, (size_t)N * 5 * H);

  // adjacency normalization + 2-hop diagonal
  deg_kernel  <<<cdiv(E, T), T, 0, stream>>>(ei, E, deg);
  dinv_kernel <<<cdiv(N, T), T, 0, stream>>>(deg, N);
  wdiag_kernel<<<cdiv(E, T), T, 0, stream>>>(ei, E, deg, wbuf, diag2);

  // embed GEMM: h = relu(x @ Wemb^T + b)   (32 rows x 64 cols per wave)
  int rb = cdiv(N, 32);                 // 3125 row blocks
  wmma_lin_kernel<2, 4><<<cdiv(rb, 8), T, 0, stream>>>(
      x, IN, Wemb, IN, bemb, hbuf, H, N, rb, IN, /*relu=*/1, H);

  // BatchNorm (training-mode batch stats)
  bn_stats_kernel<<<512, T, 0, stream>>>(hbuf, N, sums, sumsq);
  bn_final_kernel<<<1, 64, 0, stream>>>(sums, sumsq, gamma, beta, N, scsh);
  bn_apply_kernel<<<cdiv((long)N * H, T), T, 0, stream>>>(hbuf, scsh, z, cur, N);

  // hops: slots in z are [h | h1a | h2a | h1b | h2b] at col offsets 0,64,128,192,256
  int sb = cdiv(E, 8);                  // 8 waves (edges) per block
  spmm_kernel<<<sb, T, 0, stream>>>(ei, wbuf, E, cur,      H,   z + 64,  5 * H);
  spmm_kernel<<<sb, T, 0, stream>>>(ei, wbuf, E, z + 64, 5 * H, z + 128, 5 * H);
  h2fix_kernel<<<cdiv((long)N * H, T), T, 0, stream>>>(z, cur, diag2, 64, 128, N);
  spmm_kernel<<<sb, T, 0, stream>>>(ei, wbuf, E, cur,      H,   z + 192, 5 * H);
  spmm_kernel<<<sb, T, 0, stream>>>(ei, wbuf, E, z + 192, 5 * H, z + 256, 5 * H);
  h2fix_kernel<<<cdiv((long)N * H, T), T, 0, stream>>>(z, cur, diag2, 192, 256, N);

  // g = relu(z @ W1^T + b1)  (g reuses cur buffer; GEMM reads only z)
  wmma_lin_kernel<2, 4><<<cdiv(rb, 8), T, 0, stream>>>(
      z, 5 * H, W1, 5 * H, b1, cur, H, N, rb, 5 * H, /*relu=*/1, H);

  // out = g @ W2^T + b2   (C=40 -> 3 col tiles, guarded)
  wmma_lin_kernel<2, 3><<<cdiv(rb, 8), T, 0, stream>>>(
      cur, H, W2, H, b2, (float*)d_out, C, N, rb, H, /*relu=*/0, C);
}